// LowPass_63239098466970
// MI455X (gfx1250) — compile-verified
//
#include <hip/hip_runtime.h>

typedef __attribute__((ext_vector_type(2))) float v2f;
typedef __attribute__((ext_vector_type(4))) float v4f;
typedef __attribute__((ext_vector_type(8))) float v8f;

#define SEQ_T          262144
#define NBATCH         64
#define LTAP           64      // truncated IIR impulse length; |pole|^48 ~ 4e-10 << f32 eps
#define CTILES         4       // history tiles of 16 samples
#define TILES_PER_WAVE 16      // 256 output samples per wave
#define SEGS           (SEQ_T / (16 * TILES_PER_WAVE))   // 1024

// ---------------- compile-time filter tables ----------------
// Coefficients are fixed by the reference (cutoff = 0.8*Nyquist, Q = 0.707).
// w0 = 0.8*pi ; cos(w0) = -(1+sqrt(5))/4 ; sin(w0) = sin(144 deg)
struct alignas(16) Tables { float G[CTILES][16][16]; };

constexpr Tables make_tables() {
  const double cosw = -0.80901699437494742410229341718282;
  const double sinw =  0.58778525229247312916870595463907;
  const double q    = 0.707;
  const double alpha = sinw / (2.0 * q);
  const double a0d   = 1.0 + alpha;
  // round to f32 exactly as the reference does
  const float b0f = (float)((1.0 - cosw) / 2.0 / a0d);
  const float b1f = (float)((1.0 - cosw) / a0d);
  const float b2f = b0f;
  const float a1f = (float)(-2.0 * cosw / a0d);
  const float a2f = (float)((1.0 - alpha) / a0d);

  // IIR-only impulse response h (shifted by 2 so h[-1], h[-2] = 0)
  double hs[LTAP + 2] = {};
  for (int j = 0; j < LTAP; ++j)
    hs[j + 2] = (j == 0 ? 1.0 : 0.0)
              - (double)a1f * hs[j + 1] - (double)a2f * hs[j];
  // full biquad impulse response g = FIR(b) applied to h
  double g[LTAP] = {};
  for (int j = 0; j < LTAP; ++j)
    g[j] = (double)b0f * hs[j + 2] + (double)b1f * hs[j + 1] + (double)b2f * hs[j];

  // Toeplitz operator tiles: y[t0+m] = sum_c sum_k G[c][m][k] * x[t0 - 16c + k]
  Tables t = {};
  for (int c = 0; c < CTILES; ++c)
    for (int m = 0; m < 16; ++m)
      for (int k = 0; k < 16; ++k) {
        const int j = m + 16 * c - k;
        t.G[c][m][k] = (j >= 0 && j < LTAP) ? (float)g[j] : 0.0f;
      }
  return t;
}

__device__ __constant__ Tables g_tab = make_tables();

// ---------------- kernel ----------------
// Wave layout (wave32): lanes l and l+16 both serve batch column n = l&15.
// A (16x4 f32, M x K): VGPR0 = K {0 | 2}, VGPR1 = K {1 | 3} across lane halves.
// B (4x16 f32, K x N): mirrored striping -> per lane a contiguous K pair.
// D (16x16 f32): VGPR r = time row r (lanes 0-15) / r+8 (lanes 16-31), N = lane&15.
__global__ __launch_bounds__(256) void lowpass_wmma_kernel(
    const float* __restrict__ x, float* __restrict__ y) {
  const int tid  = blockIdx.x * blockDim.x + threadIdx.x;
  const int lane = threadIdx.x & 31;
  const int wave = tid >> 5;
  const int bg   = wave & 3;           // batch group: 16 rows
  const int seg  = wave >> 2;          // time segment: 256 samples
  const int m    = lane & 15;          // matrix row (A) / column n (B, D)
  const int half = (lane >> 4) & 1;    // lane-half -> K pair {0,1} vs {2,3}
  const int koff = half * 2;

  const int n = bg * 16 + m;           // batch row owned by this lane
  const float* __restrict__ row  = x + (size_t)n * SEQ_T;
  float*       __restrict__ orow = y + (size_t)n * SEQ_T;
  const int t_start = seg * (16 * TILES_PER_WAVE);

  // Resident A operands: constant Toeplitz tiles G_c, chunked K=4.
  v2f A[CTILES][4];
#pragma unroll
  for (int c = 0; c < CTILES; ++c)
#pragma unroll
    for (int kk = 0; kk < 4; ++kk)
      A[c][kk] = *(const v2f*)&g_tab.G[c][m][4 * kk + koff];

  // Input-tile register ring: slot (j & 3) holds input tile at t_start + 16*j.
  // All three history slots are in-range iff seg != 0 (scalar, wave-uniform
  // branch -> EXEC stays all-ones, as required around WMMA).
  v2f Xr[4][4];
  if (seg != 0) {
#pragma unroll
    for (int j = -3; j < 0; ++j) {
      const int t0c = t_start + 16 * j;
      const int s   = j & 3;
#pragma unroll
      for (int kk = 0; kk < 4; ++kk)
        Xr[s][kk] = *(const v2f*)(row + t0c + 4 * kk + koff);
    }
  } else {
    const v2f vzero = {0.0f, 0.0f};     // x[t<0] = 0 (zero initial state)
#pragma unroll
    for (int j = -3; j < 0; ++j) {
      const int s = j & 3;
#pragma unroll
      for (int kk = 0; kk < 4; ++kk)
        Xr[s][kk] = vzero;
    }
  }

#pragma unroll
  for (int i = 0; i < TILES_PER_WAVE; ++i) {
    const int t0 = t_start + 16 * i;

    // Load newest input tile into the ring (8B-aligned b64 per lane).
    const int s0 = i & 3;
#pragma unroll
    for (int kk = 0; kk < 4; ++kk)
      Xr[s0][kk] = *(const v2f*)(row + t0 + 4 * kk + koff);

    // Y_tile = sum_c G_c * X(t0 - 16c), via chained 16x16x4 f32 WMMA.
    v8f acc = {};
#pragma unroll
    for (int c = 0; c < CTILES; ++c) {
      const int s = (i - c) & 3;
#pragma unroll
      for (int kk = 0; kk < 4; ++kk)
        acc = __builtin_amdgcn_wmma_f32_16x16x4_f32(
            false, A[c][kk], false, Xr[s][kk], (short)0, acc, false, false);
    }

    // Clamp to [-1, 1] (torchaudio lfilter default) and store.
    v4f lo, hi;
#pragma unroll
    for (int r = 0; r < 4; ++r) {
      lo[r] = fminf(1.0f, fmaxf(-1.0f, acc[r]));
      hi[r] = fminf(1.0f, fmaxf(-1.0f, acc[r + 4]));
    }
    // Lane owns times t0 + half*8 + {0..7}: two aligned b128 stores.
    // Output is never re-read by this kernel -> non-temporal (spare L2 for x).
    __builtin_nontemporal_store(lo, (v4f*)(orow + t0 + half * 8));
    __builtin_nontemporal_store(hi, (v4f*)(orow + t0 + half * 8 + 4));
  }
}

extern "C" void kernel_launch(void* const* d_in, const int* in_sizes, int n_in,
                              void* d_out, int out_size, void* d_ws, size_t ws_size,
                              hipStream_t stream) {
  (void)in_sizes; (void)n_in; (void)out_size; (void)d_ws; (void)ws_size;
  const float* xin = (const float*)d_in[0];
  float* yout = (float*)d_out;
  // 4 batch groups x 1024 segments = 4096 waves = 131072 threads = 512 blocks
  const int waves  = 4 * SEGS;
  const int blocks = (waves * 32) / 256;
  lowpass_wmma_kernel<<<blocks, 256, 0, stream>>>(xin, yout);
}